// PathRerankerGNN_81149112090941
// MI455X (gfx1250) — compile-verified
//
#include <hip/hip_runtime.h>
#include <hip/hip_bf16.h>

#define NN 50000
#define EE 1600000
#define FEAT 256
#define HID 128

typedef float v2f __attribute__((ext_vector_type(2)));
typedef float v8f __attribute__((ext_vector_type(8)));
typedef int   v4i_vs __attribute__((vector_size(16)));   // matches builtin param

// Async global->LDS staging (GLOBAL_LOAD_ASYNC_TO_LDS_B128, ASYNCcnt) if the
// toolchain declares the builtins; otherwise fall back to load/ds_store.
#if defined(__has_builtin)
# if __has_builtin(__builtin_amdgcn_global_load_async_to_lds_b128) && \
     __has_builtin(__builtin_amdgcn_s_wait_asynccnt)
#  define HAVE_ASYNC_LDS 1
# endif
#endif
#ifndef HAVE_ASYNC_LDS
# define HAVE_ASYNC_LDS 0
#endif

// Stage NF floats (16-row tile) into LDS with 256 threads, then barrier.
template <int NF>
__device__ inline void stage_tile(float* As, const float* __restrict__ Arow,
                                  int tid) {
#if HAVE_ASYNC_LDS
    auto* gp = (__attribute__((address_space(1))) v4i_vs*)((void*)Arow);
    auto* lp = (__attribute__((address_space(3))) v4i_vs*)((void*)As);
#pragma unroll
    for (int i = tid; i < NF / 4; i += 256)   // 16B chunks
        __builtin_amdgcn_global_load_async_to_lds_b128(gp + i, lp + i, 0, 0);
    __builtin_amdgcn_s_wait_asynccnt(0);      // own async loads landed in LDS
#else
#pragma unroll
    for (int i = tid; i < NF; i += 256) As[i] = Arow[i];
#endif
    __syncthreads();                          // all waves' tiles visible
}

// ---------------------------------------------------------------------------
// WMMA mainloop: one wave computes a 16x16 f32 tile of  A(16xK) @ B(Kx128)
// using V_WMMA_F32_16X16X4_F32.  A staged in LDS row-major [16][K]; B read
// from global (row-major [K][128], hot in L2 across all row-blocks).
// f32 A-frag layout (ISA 7.12.2): lanes 0-15 -> M=0..15 with K=k+0/k+1 in
// vgpr0/vgpr1; lanes 16-31 -> K=k+2/k+3.  B mirrors with N in place of M.
// ---------------------------------------------------------------------------
template <int K>
__device__ inline v8f wmma_rowtile(const float* __restrict__ B,
                                   const float* __restrict__ As,
                                   int wave, int lane) {
    const int lh = lane & 15;
    const int kh = (lane >> 4) << 1;          // 0 or 2
    const float* bp = B + wave * 16 + lh;     // column slice for this wave
    v8f acc = {};
#pragma unroll 4
    for (int k = 0; k < K; k += 4) {
        v2f a, b;
        a.x = As[lh * K + k + kh];
        a.y = As[lh * K + k + kh + 1];
        b.x = bp[(size_t)(k + kh) * HID];
        b.y = bp[(size_t)(k + kh + 1) * HID];
        acc = __builtin_amdgcn_wmma_f32_16x16x4_f32(
            false, a, false, b, (short)0, acc, false, false);
    }
    return acc;
}

__device__ inline float gelu_erf(float x) {
    return 0.5f * x * (1.0f + erff(x * 0.70710678118654752f));
}

// ---- kernel 1: h = gelu(layernorm(x @ W_in + b_in)) -----------------------
__global__ __launch_bounds__(256) void k_inproj(
    const float* __restrict__ x, const float* __restrict__ W,
    const float* __restrict__ bias, const float* __restrict__ g,
    const float* __restrict__ bvec, float* __restrict__ h) {
    __shared__ __align__(16) float As[16 * FEAT];   // 16 KB
    __shared__ __align__(16) float Cs[16 * HID];    //  8 KB
    const int tid = threadIdx.x;
    const int br  = blockIdx.x;

    stage_tile<16 * FEAT>(As, x + (size_t)br * 16 * FEAT, tid);

    const int wave = tid >> 5, lane = tid & 31;
    v8f acc = wmma_rowtile<FEAT>(W, As, wave, lane);

    const int lh  = lane & 15;
    const int mb  = (lane < 16) ? 0 : 8;       // C layout: vgpr i -> M=i(+8)
    const int col = wave * 16 + lh;
    const float bb = bias[col];
#pragma unroll
    for (int i = 0; i < 8; ++i) Cs[(mb + i) * HID + col] = acc[i] + bb;
    __syncthreads();

    // LayerNorm + GELU: 16 threads per row, 8 cols each
    const int r = tid >> 4, c0 = tid & 15;
    float vals[8], s = 0.f, s2 = 0.f;
#pragma unroll
    for (int j = 0; j < 8; ++j) {
        float v = Cs[r * HID + c0 + 16 * j];
        vals[j] = v; s += v; s2 += v * v;
    }
#pragma unroll
    for (int off = 8; off >= 1; off >>= 1) {
        s  += __shfl_xor(s, off, 32);
        s2 += __shfl_xor(s2, off, 32);
    }
    const float mean = s * (1.0f / HID);
    const float var  = s2 * (1.0f / HID) - mean * mean;
    const float rstd = rsqrtf(var + 1e-5f);
    const size_t rowoff = ((size_t)br * 16 + r) * HID;
#pragma unroll
    for (int j = 0; j < 8; ++j) {
        int c = c0 + 16 * j;
        float y = (vals[j] - mean) * rstd * g[c] + bvec[c];
        h[rowoff + c] = gelu_erf(y);
    }
}

// ---- kernel 2: hW = h @ W;  s_src = hW@a_src,  s_dst = hW@a_dst ----------
__global__ __launch_bounds__(256) void k_gat_proj(
    const float* __restrict__ h, const float* __restrict__ W,
    const float* __restrict__ asrc, const float* __restrict__ adst,
    float* __restrict__ hW, float* __restrict__ ssrc, float* __restrict__ sdst) {
    __shared__ __align__(16) float As[16 * HID];
    __shared__ __align__(16) float Cs[16 * HID];
    const int tid = threadIdx.x;
    const int br  = blockIdx.x;

    stage_tile<16 * HID>(As, h + (size_t)br * 16 * HID, tid);

    const int wave = tid >> 5, lane = tid & 31;
    v8f acc = wmma_rowtile<HID>(W, As, wave, lane);

    const int lh  = lane & 15;
    const int mb  = (lane < 16) ? 0 : 8;
    const int col = wave * 16 + lh;
#pragma unroll
    for (int i = 0; i < 8; ++i) {
        float v = acc[i];
        Cs[(mb + i) * HID + col] = v;
        hW[((size_t)br * 16 + mb + i) * HID + col] = v;
    }
    __syncthreads();

    const int r = tid >> 4, c0 = tid & 15;
    float d1 = 0.f, d2 = 0.f;
#pragma unroll
    for (int j = 0; j < 8; ++j) {
        int c = c0 + 16 * j;
        float v = Cs[r * HID + c];
        d1 += v * asrc[c];
        d2 += v * adst[c];
    }
#pragma unroll
    for (int off = 8; off >= 1; off >>= 1) {
        d1 += __shfl_xor(d1, off, 32);
        d2 += __shfl_xor(d2, off, 32);
    }
    if (c0 == 0) {
        ssrc[(size_t)br * 16 + r] = d1;
        sdst[(size_t)br * 16 + r] = d2;
    }
}

// ---- elementwise zero -----------------------------------------------------
__global__ __launch_bounds__(256) void k_zero(float* __restrict__ p, size_t n) {
    size_t i = (size_t)blockIdx.x * 256 + threadIdx.x;
    if (i < n) p[i] = 0.0f;
}

// ---- edge pass 1: alpha + segment-max (bitwise uint atomicMax, m >= 0) ----
__global__ __launch_bounds__(256) void k_edge_alpha(
    const int* __restrict__ src, const int* __restrict__ dst,
    const float* __restrict__ ssrc, const float* __restrict__ sdst,
    float* __restrict__ alpha, unsigned int* __restrict__ mbits) {
    int e = blockIdx.x * 256 + threadIdx.x;
    if (e >= EE) return;
    float sv = ssrc[src[e]] + sdst[dst[e]];
    float a  = sv > 0.0f ? sv : 0.2f * sv;     // leaky_relu(0.2)
    alpha[e] = a;
    if (a > 0.0f) atomicMax(mbits + dst[e], __float_as_uint(a));
}

// ---- node: em = exp(-m), denom = E * em  (zero-scatter softmax floor) -----
__global__ __launch_bounds__(256) void k_node_em(
    const float* __restrict__ m, float* __restrict__ em, float* __restrict__ denom) {
    int n = blockIdx.x * 256 + threadIdx.x;
    if (n >= NN) return;
    float e_ = expf(-m[n]);
    em[n]    = e_;
    denom[n] = (float)EE * e_;
}

// ---- edge pass 2: ex = exp(a - m[dst]); denom[dst] += ex - em[dst] --------
__global__ __launch_bounds__(256) void k_edge_denom(
    const int* __restrict__ dst, float* __restrict__ alpha,
    const float* __restrict__ m, const float* __restrict__ em,
    float* __restrict__ denom) {
    int e = blockIdx.x * 256 + threadIdx.x;
    if (e >= EE) return;
    int d = dst[e];
    float ex = expf(alpha[e] - m[d]);
    alpha[e] = ex;
    atomicAdd(denom + d, ex - em[d]);
}

// ---- edge pass 3: one wave per edge; agg[dst] += (ex/denom[dst])*hW[src] --
__global__ __launch_bounds__(256) void k_edge_agg(
    const int* __restrict__ src, const int* __restrict__ dst,
    const float* __restrict__ ex, const float* __restrict__ denom,
    const float* __restrict__ hW, float* __restrict__ agg) {
    int wv   = (blockIdx.x * 256 + threadIdx.x) >> 5;
    int lane = threadIdx.x & 31;
    if (wv >= EE) return;
    int s = src[wv], d = dst[wv];
    float w = ex[wv] / denom[d];
    const float4* hp = (const float4*)(hW + (size_t)s * HID);
    float4 v = hp[lane];                       // 32 lanes x 16B = 128 floats
    float* ap = agg + (size_t)d * HID + lane * 4;
    atomicAdd(ap + 0, w * v.x);
    atomicAdd(ap + 1, w * v.y);
    atomicAdd(ap + 2, w * v.z);
    atomicAdd(ap + 3, w * v.w);
}

// ---- node update: h = layernorm(h + elu(agg)) (in place) ------------------
__global__ __launch_bounds__(128) void k_node_update(
    float* __restrict__ h, const float* __restrict__ agg,
    const float* __restrict__ g, const float* __restrict__ b) {
    const int n = blockIdx.x, c = threadIdx.x;
    const size_t i = (size_t)n * HID + c;
    float a = agg[i];
    a = a > 0.0f ? a : expm1f(a);              // elu
    float v = h[i] + a;

    float s = v, s2 = v * v;
#pragma unroll
    for (int off = 16; off >= 1; off >>= 1) {
        s  += __shfl_xor(s, off, 32);
        s2 += __shfl_xor(s2, off, 32);
    }
    __shared__ float ps[4], ps2[4];
    const int wid = c >> 5, lane = c & 31;
    if (lane == 0) { ps[wid] = s; ps2[wid] = s2; }
    __syncthreads();
    s  = ps[0] + ps[1] + ps[2] + ps[3];
    s2 = ps2[0] + ps2[1] + ps2[2] + ps2[3];
    const float mean = s * (1.0f / HID);
    const float var  = s2 * (1.0f / HID) - mean * mean;
    const float rstd = rsqrtf(var + 1e-5f);
    h[i] = (v - mean) * rstd * g[c] + b[c];
}

// ---- global mean pool (partial column sums + atomics) ---------------------
__global__ __launch_bounds__(128) void k_pool(
    const float* __restrict__ h, float* __restrict__ pooled) {
    const int c = threadIdx.x;
    float s = 0.0f;
    for (int n = blockIdx.x; n < NN; n += gridDim.x)
        s += h[(size_t)n * HID + c];
    atomicAdd(pooled + c, s);
}

// ---- score head: gelu(g@Ws1+bs1) @ Ws2 + bs2 ------------------------------
__global__ __launch_bounds__(128) void k_final(
    const float* __restrict__ pooled, const float* __restrict__ Ws1,
    const float* __restrict__ bs1, const float* __restrict__ Ws2,
    const float* __restrict__ bs2, float* __restrict__ out) {
    __shared__ float gm[HID];
    __shared__ float hid[HID / 2];
    const int t = threadIdx.x;
    gm[t] = pooled[t] * (1.0f / NN);
    __syncthreads();
    if (t < HID / 2) {
        float s = bs1[t];
        for (int c = 0; c < HID; ++c) s += gm[c] * Ws1[c * (HID / 2) + t];
        hid[t] = gelu_erf(s);
    }
    __syncthreads();
    if (t == 0) {
        float s = bs2[0];
        for (int j = 0; j < HID / 2; ++j) s += hid[j] * Ws2[j];
        out[0] = s;
    }
}

// ---------------------------------------------------------------------------
extern "C" void kernel_launch(void* const* d_in, const int* in_sizes, int n_in,
                              void* d_out, int out_size, void* d_ws, size_t ws_size,
                              hipStream_t stream) {
    const float* x      = (const float*)d_in[0];
    const int*   eidx   = (const int*)d_in[1];
    const float* W_in   = (const float*)d_in[3];
    const float* b_in   = (const float*)d_in[4];
    const float* ln0_g  = (const float*)d_in[5];
    const float* ln0_b  = (const float*)d_in[6];
    const float* Ws1    = (const float*)d_in[18];
    const float* bs1    = (const float*)d_in[19];
    const float* Ws2    = (const float*)d_in[20];
    const float* bs2    = (const float*)d_in[21];

    const int* src = eidx;          // edge_index[0]
    const int* dst = eidx + EE;     // edge_index[1]

    // workspace layout (floats)
    const size_t NH = (size_t)NN * HID;
    float* ws     = (float*)d_ws;
    float* h      = ws;             // NH
    float* hW     = h + NH;         // NH
    float* agg    = hW + NH;        // NH
    float* alpha  = agg + NH;       // EE
    float* m      = alpha + EE;     // NN
    float* em     = m + NN;         // NN
    float* denom  = em + NN;        // NN
    float* ssrc   = denom + NN;     // NN
    float* sdst   = ssrc + NN;      // NN
    float* pooled = sdst + NN;      // HID
    const size_t need = (size_t)(pooled + HID - ws) * sizeof(float);
    if (ws_size < need) return;     // insufficient scratch (deterministic no-op)

    const int RB  = NN / 16;                 // 3125 row blocks (N % 16 == 0)
    const int EB  = (EE + 255) / 256;        // edge-thread blocks
    const int EWB = EE / 8;                  // edge-wave blocks (256 thr = 8 waves)

    // 1) input projection + LN + GELU
    k_inproj<<<RB, 256, 0, stream>>>(x, W_in, b_in, ln0_g, ln0_b, h);

    // 2) two GAT layers
    for (int L = 0; L < 2; ++L) {
        const float* W  = (const float*)d_in[L ? 13 : 8];
        const float* as = (const float*)d_in[L ? 14 : 9];
        const float* ad = (const float*)d_in[L ? 15 : 10];
        const float* lg = (const float*)d_in[L ? 16 : 11];
        const float* lb = (const float*)d_in[L ? 17 : 12];

        k_gat_proj<<<RB, 256, 0, stream>>>(h, W, as, ad, hW, ssrc, sdst);
        k_zero<<<(NN + 255) / 256, 256, 0, stream>>>(m, (size_t)NN);
        k_edge_alpha<<<EB, 256, 0, stream>>>(src, dst, ssrc, sdst, alpha,
                                             (unsigned int*)m);
        k_node_em<<<(NN + 255) / 256, 256, 0, stream>>>(m, em, denom);
        k_zero<<<(unsigned)((NH + 255) / 256), 256, 0, stream>>>(agg, NH);
        k_edge_denom<<<EB, 256, 0, stream>>>(dst, alpha, m, em, denom);
        k_edge_agg<<<EWB, 256, 0, stream>>>(src, dst, alpha, denom, hW, agg);
        k_node_update<<<NN, 128, 0, stream>>>(h, agg, lg, lb);
    }

    // 3) mean pool + score MLP
    k_zero<<<1, 256, 0, stream>>>(pooled, (size_t)HID);
    k_pool<<<512, 128, 0, stream>>>(h, pooled);
    k_final<<<1, 128, 0, stream>>>(pooled, Ws1, bs1, Ws2, bs2, (float*)d_out);
}